// HGCN_73203422593575
// MI455X (gfx1250) — compile-verified
//
#include <hip/hip_runtime.h>
#include <math.h>

#define EPSF 1e-8f

typedef __attribute__((ext_vector_type(2))) float v2f;
typedef __attribute__((ext_vector_type(8))) float v8f;

// ---------------------------------------------------------------------------
// Cross-ratio of rows 0..3 of M [4 x 64] under the UHG bilinear form, and
// (optionally) the guarded restore-cross-ratio scale factor.
// One wave of 32 lanes; lane i covers columns i and i+32 (col 63 weight -1).
// ---------------------------------------------------------------------------
__global__ void k_cr_scale(const float* __restrict__ M,
                           const float* __restrict__ crPrev,
                           float* __restrict__ crOut,
                           float* __restrict__ scaleOut)
{
    const int lane = threadIdx.x;                       // 32 threads
    float a0 = M[0*64 + lane], a1 = M[0*64 + lane + 32];
    float b0 = M[1*64 + lane], b1 = M[1*64 + lane + 32];
    float c0 = M[2*64 + lane], c1 = M[2*64 + lane + 32];
    float d0 = M[3*64 + lane], d1 = M[3*64 + lane + 32];
    float w  = (lane == 31) ? -1.0f : 1.0f;             // column 63 gets -1
    float ac = a0*c0 + w*a1*c1;
    float bd = b0*d0 + w*b1*d1;
    float ad = a0*d0 + w*a1*d1;
    float bc = b0*c0 + w*b1*c1;
#pragma unroll
    for (int m = 16; m >= 1; m >>= 1) {
        ac += __shfl_xor(ac, m, 32);
        bd += __shfl_xor(bd, m, 32);
        ad += __shfl_xor(ad, m, 32);
        bc += __shfl_xor(bc, m, 32);
    }
    if (lane == 0) {
        float cr = (ac * bd) / (ad * bc + EPSF);
        if (crOut) crOut[0] = cr;
        if (scaleOut) {
            float prev  = crPrev[0];
            float ratio = prev / (cr + EPSF);
            bool cond = (cr == cr) && (prev == prev) &&
                        (fabsf(cr) > EPSF) && (fabsf(prev) > EPSF) &&
                        (ratio > EPSF);
            float sc = cond ? sqrtf(fabsf(ratio)) : 1.0f;
            bool ok  = (sc == sc) && (fabsf(sc) <= 3.0e38f); // inputs are normalized => finite
            scaleOut[0] = (cond && ok) ? sc : 1.0f;
        }
    }
}

// ---------------------------------------------------------------------------
// y = proj_normalize( (preScale * X/max(cnt,1)) @ W + b ), optional ReLU.
// One wave per 16-row tile; 4 waves per block. f32 WMMA 16x16x4:
//   A frag: lanes 0-15 -> (K=k,k+1) of row r0+lane, lanes 16-31 -> (K=k+2,k+3)
//   B frag: W staged transposed in LDS (stride 68 -> conflict-free b64 reads)
//   C/D:    vgpr v = row v (lanes 0-15) / row v+8 (lanes 16-31), col = lane&15
// ---------------------------------------------------------------------------
__global__ void __launch_bounds__(128)
k_hyp_linear(const float* __restrict__ X, const float* __restrict__ Wmat,
             const float* __restrict__ bias, float* __restrict__ Y,
             const float* __restrict__ preScale, const float* __restrict__ cnt,
             int nRows, int doRelu)
{
    __shared__ float Wl[64 * 68];          // W transposed: Wl[n*68 + k] = W[k][n]
    const int tid  = threadIdx.x;
    const int lane = tid & 31;
    const int wave = tid >> 5;

    for (int i = tid; i < 64 * 64; i += 128) {
        int k = i >> 6, n = i & 63;
        Wl[n * 68 + k] = Wmat[i];
    }
    __syncthreads();

    const int tile = blockIdx.x * 4 + wave;
    if (tile * 16 >= nRows) return;        // wave-uniform; EXEC stays all-ones
    const int r0 = tile * 16;
    const int nl = lane & 15;              // A-row-in-tile / B,C,D col-in-tile
    const int hi = lane >> 4;              // K-half for A/B, M-half for C/D
    int arow = r0 + nl;
    if (arow >= nRows) arow = nRows - 1;   // clamp (N % 16 == 0 in practice)

    float s = 1.0f;
    if (preScale) s = preScale[0];
    if (cnt)      s = s / fmaxf(cnt[arow], 1.0f);

    const float* xrow = X + (size_t)arow * 64;
    v8f acc0 = {}, acc1 = {}, acc2 = {}, acc3 = {};

#pragma unroll
    for (int k = 0; k < 64; k += 4) {
        const int kk = k + hi * 2;
        v2f a = *(const v2f*)(xrow + kk);
        a.x *= s; a.y *= s;
        v2f bb0 = *(const v2f*)(&Wl[( 0 + nl) * 68 + kk]);
        v2f bb1 = *(const v2f*)(&Wl[(16 + nl) * 68 + kk]);
        v2f bb2 = *(const v2f*)(&Wl[(32 + nl) * 68 + kk]);
        v2f bb3 = *(const v2f*)(&Wl[(48 + nl) * 68 + kk]);
        acc0 = __builtin_amdgcn_wmma_f32_16x16x4_f32(false, a, false, bb0, (short)0, acc0, false, false);
        acc1 = __builtin_amdgcn_wmma_f32_16x16x4_f32(false, a, false, bb1, (short)0, acc1, false, false);
        acc2 = __builtin_amdgcn_wmma_f32_16x16x4_f32(false, a, false, bb2, (short)0, acc2, false, false);
        acc3 = __builtin_amdgcn_wmma_f32_16x16x4_f32(false, a, false, bb3, (short)0, acc3, false, false);
    }

    const float bnt0 = bias[ 0 + nl];
    const float bnt1 = bias[16 + nl];
    const float bnt2 = bias[32 + nl];
    const float bnt3 = bias[48 + nl];

#pragma unroll
    for (int v = 0; v < 8; v++) {
        float y0 = acc0[v] + bnt0;
        float y1 = acc1[v] + bnt1;
        float y2 = acc2[v] + bnt2;
        float y3 = acc3[v] + bnt3;
        float p = y0*y0 + y1*y1 + y2*y2 + y3*y3;
        p += __shfl_xor(p, 1, 32);         // reduce across 16-lane half: full row norm
        p += __shfl_xor(p, 2, 32);
        p += __shfl_xor(p, 4, 32);
        p += __shfl_xor(p, 8, 32);
        float inv = 1.0f / (sqrtf(p) + EPSF);
        y0 *= inv; y1 *= inv; y2 *= inv; y3 *= inv;
        if (doRelu) {
            y0 = fmaxf(y0, 0.0f); y1 = fmaxf(y1, 0.0f);
            y2 = fmaxf(y2, 0.0f); y3 = fmaxf(y3, 0.0f);
        }
        const int orow = r0 + v + hi * 8;
        if (orow < nRows) {
            float* yp = Y + (size_t)orow * 64 + nl;
            yp[0] = y0; yp[16] = y1; yp[32] = y2; yp[48] = y3;
        }
    }
}

// ---------------------------------------------------------------------------
__global__ void k_zero(float* __restrict__ p, long long n)
{
    long long i = (long long)blockIdx.x * blockDim.x + threadIdx.x;
    const long long stride = (long long)gridDim.x * blockDim.x;
    for (; i < n; i += stride) p[i] = 0.0f;
}

// scatter-add: 16 threads per edge, float4 per thread; thread q==0 bumps count
__global__ void k_scatter(const float* __restrict__ H, const int* __restrict__ src,
                          const int* __restrict__ dst, float* __restrict__ summed,
                          float* __restrict__ cnt, int E)
{
    const long long t = (long long)blockIdx.x * blockDim.x + threadIdx.x;
    if (t >= (long long)E * 16) return;
    const int e = (int)(t >> 4);
    const int q = (int)(t & 15);
    const int s = src[e];
    const int d = dst[e];
    const float4 v = *(const float4*)(H + (size_t)s * 64 + q * 4);
    float* o = summed + (size_t)d * 64 + q * 4;
    atomicAdd(o + 0, v.x);
    atomicAdd(o + 1, v.y);
    atomicAdd(o + 2, v.z);
    atomicAdd(o + 3, v.w);
    if (q == 0) atomicAdd(&cnt[d], 1.0f);
}

__global__ void k_scale4(float* __restrict__ p, const float* __restrict__ sPtr, long long n4)
{
    const long long i = (long long)blockIdx.x * blockDim.x + threadIdx.x;
    if (i >= n4) return;
    const float s = sPtr[0];
    float4 v = ((const float4*)p)[i];
    v.x *= s; v.y *= s; v.z *= s; v.w *= s;
    ((float4*)p)[i] = v;
}

// ---------------------------------------------------------------------------
extern "C" void kernel_launch(void* const* d_in, const int* in_sizes, int n_in,
                              void* d_out, int out_size, void* d_ws, size_t ws_size,
                              hipStream_t stream)
{
    const float* x  = (const float*)d_in[0];
    const int*   ei = (const int*)d_in[1];
    const float* W0 = (const float*)d_in[2];
    const float* b0 = (const float*)d_in[3];
    const float* Wm = (const float*)d_in[4];
    const float* bm = (const float*)d_in[5];
    const float* W1 = (const float*)d_in[6];
    const float* b1 = (const float*)d_in[7];
    const int N = in_sizes[0] / 64;
    const int E = in_sizes[1] / 2;
    const int* src = ei;
    const int* dst = ei + E;

    float* out    = (float*)d_out;
    float* h0     = out;                        // stage h0 in d_out (overwritten later)
    float* summed = (float*)d_ws;               // N*64 (becomes h1 in place)
    float* cnt    = summed + (size_t)N * 64;    // N
    float* scal   = cnt + N;                    // [cr_init, cr0, scale0, scale1]

    const int tiles = (N + 15) / 16;
    const int gemmBlocks = (tiles + 3) / 4;

    // cr_init from x
    k_cr_scale<<<1, 32, 0, stream>>>(x, nullptr, &scal[0], nullptr);
    // h0 = normalize(x @ W0 + b0)
    k_hyp_linear<<<gemmBlocks, 128, 0, stream>>>(x, W0, b0, h0, nullptr, nullptr, N, 0);
    // cr0 from h0
    k_cr_scale<<<1, 32, 0, stream>>>(h0, nullptr, &scal[1], nullptr);
    // zero accumulators
    k_zero<<<1024, 256, 0, stream>>>(summed, (long long)N * 64 + N);
    // scatter-mean numerator + counts
    const long long st = (long long)E * 16;
    k_scatter<<<(int)((st + 255) / 256), 256, 0, stream>>>(h0, src, dst, summed, cnt, E);
    // h1 = normalize((summed/max(cnt,1)) @ Wm + bm)   (in place over summed)
    k_hyp_linear<<<gemmBlocks, 128, 0, stream>>>(summed, Wm, bm, summed, nullptr, cnt, N, 0);
    // scale0: restore h1 toward cr0
    k_cr_scale<<<1, 32, 0, stream>>>(summed, &scal[1], nullptr, &scal[2]);
    // h2 = relu(normalize((scale0*h1) @ W1 + b1)) -> d_out
    k_hyp_linear<<<gemmBlocks, 128, 0, stream>>>(summed, W1, b1, out, &scal[2], nullptr, N, 1);
    // scale1: restore h2 toward cr_init
    k_cr_scale<<<1, 32, 0, stream>>>(out, &scal[0], nullptr, &scal[3]);
    // out *= scale1
    const long long n4 = (long long)N * 16;
    k_scale4<<<(int)((n4 + 255) / 256), 256, 0, stream>>>(out, &scal[3], n4);
}